// Coulomb_24610162606257
// MI455X (gfx1250) — compile-verified
//
#include <hip/hip_runtime.h>
#include <hip/hip_bf16.h>

// ---------------------------------------------------------------------------
// Coulomb pair energy, MI455X (gfx1250, wave32)
//
// HBM traffic = pairs stream only (64 MB, NT b128 loads). coords/charges are
// L2-resident gathers. Reduction: wave-level via V_WMMA_F32_16X16X4_F32
// (B = ones => D[m][n] = p[m] + p[m+16], full f32 precision), then LDS across
// the 8 waves of the block, then a deterministic single-block finalize pass.
// ---------------------------------------------------------------------------

typedef __attribute__((ext_vector_type(2))) float v2f;
typedef __attribute__((ext_vector_type(8))) float v8f;
typedef __attribute__((ext_vector_type(4))) int   v4i;

#define TPB      256
#define NBLOCKS  2048

__device__ __forceinline__ float fast_rsqrt(float x) {
#if __has_builtin(__builtin_amdgcn_rsqf)
  return __builtin_amdgcn_rsqf(x);     // v_rsq_f32
#else
  return rsqrtf(x);
#endif
}

// Full-precision f32 reduction of one value per lane across a wave32 using
// V_WMMA_F32_16X16X4_F32.  A: a[0]=p, a[1]=0 -> A[m][0]=p[lane m],
// A[m][2]=p[lane 16+m].  B = ones.  D[m][n] = p[m] + p[m+16] (all columns
// identical).  Lane L<16 holds D rows 0..7 in d[0..7]; lane L>=16 rows 8..15.
// Sum d[0..7] in-lane, then one shfl_xor(16) gives the wave total in every
// lane.  EXEC must be all ones: callers invoke this from fully-active waves.
__device__ __forceinline__ float wave_sum_wmma(float p) {
  v2f a; a[0] = p;    a[1] = 0.0f;
  v2f b; b[0] = 1.0f; b[1] = 1.0f;
  v8f c = {};
  v8f d = __builtin_amdgcn_wmma_f32_16x16x4_f32(
      /*neg_a=*/false, a, /*neg_b=*/false, b,
      /*c_mod=*/(short)0, c, /*reuse_a=*/false, /*reuse_b=*/false);
  float s = ((d[0] + d[1]) + (d[2] + d[3])) + ((d[4] + d[5]) + (d[6] + d[7]));
  s += __shfl_xor(s, 16, 32);          // combine rows 0..7 with rows 8..15
  return s;
}

__device__ __forceinline__ float pair_e(int i, int j,
                                        const float* __restrict__ coords,
                                        const float* __restrict__ charges,
                                        float Lx, float Ly, float Lz,
                                        float iLx, float iLy, float iLz,
                                        float cutoff2, float inv_cutoff) {
  float dx = coords[3 * i + 0] - coords[3 * j + 0];
  float dy = coords[3 * i + 1] - coords[3 * j + 1];
  float dz = coords[3 * i + 2] - coords[3 * j + 2];
  // minimum image (box is diagonal): s -= floor(s + 0.5)
  float sx = dx * iLx; sx -= floorf(sx + 0.5f); dx = sx * Lx;
  float sy = dy * iLy; sy -= floorf(sy + 0.5f); dy = sy * Ly;
  float sz = dz * iLz; sz -= floorf(sz + 0.5f); dz = sz * Lz;
  float r2 = dx * dx + dy * dy + dz * dz;
  float rinv = fast_rsqrt(r2);                       // r2==0 -> +inf (matches 1/0)
  float e = charges[i] * charges[j] * (rinv - inv_cutoff);
  return (r2 <= cutoff2) ? e : 0.0f;                 // mask: dr <= cutoff
}

__device__ __forceinline__ float block_reduce(float acc) {
  __shared__ float smem[TPB / 32];
  float wsum = wave_sum_wmma(acc);
  const int lane = threadIdx.x & 31;
  const int wave = threadIdx.x >> 5;
  if (lane == 0) smem[wave] = wsum;
  __syncthreads();
  float v = 0.0f;
  if (wave == 0) {
    v = (lane < (TPB / 32)) ? smem[lane] : 0.0f;
    v += __shfl_xor(v, 1, 32);
    v += __shfl_xor(v, 2, 32);
    v += __shfl_xor(v, 4, 32);
  }
  return v;  // valid in wave 0, lanes 0..7
}

__global__ void coulomb_pairs_kernel(const float* __restrict__ coords,
                                     const int*   __restrict__ pairs,   // [n_pairs][2] i32
                                     const float* __restrict__ box,     // 3x3, diagonal
                                     const float* __restrict__ charges,
                                     const int*   __restrict__ cutoff_p,
                                     float*       __restrict__ block_partials,
                                     int n_pairs) {
  const float Lx = box[0], Ly = box[4], Lz = box[8];
  const float iLx = 1.0f / Lx, iLy = 1.0f / Ly, iLz = 1.0f / Lz;
  const float cutoff = (float)(*cutoff_p);
  const float cutoff2 = cutoff * cutoff;
  const float inv_cutoff = 1.0f / cutoff;

  const int tid    = blockIdx.x * blockDim.x + threadIdx.x;
  const int stride = gridDim.x * blockDim.x;
  const int n2     = n_pairs >> 1;                 // two pairs per b128
  const v4i* __restrict__ p4 = (const v4i*)pairs;

  float acc = 0.0f;
  for (int k = tid; k < n2; k += stride) {
    __builtin_prefetch((const void*)(p4 + k + stride), 0, 0);  // global_prefetch_b8
    v4i pr = __builtin_nontemporal_load(p4 + k);               // stream-once: TH=NT
    acc += pair_e(pr[0], pr[1], coords, charges, Lx, Ly, Lz, iLx, iLy, iLz,
                  cutoff2, inv_cutoff);
    acc += pair_e(pr[2], pr[3], coords, charges, Lx, Ly, Lz, iLx, iLy, iLz,
                  cutoff2, inv_cutoff);
  }
  if ((n_pairs & 1) && tid == 0) {                 // generic odd-count tail
    acc += pair_e(pairs[2 * (n_pairs - 1)], pairs[2 * (n_pairs - 1) + 1],
                  coords, charges, Lx, Ly, Lz, iLx, iLy, iLz,
                  cutoff2, inv_cutoff);
  }

  float v = block_reduce(acc);
  if (threadIdx.x == 0) block_partials[blockIdx.x] = v;
}

__global__ void coulomb_finalize_kernel(const float* __restrict__ partials,
                                        int n,
                                        const int* __restrict__ kc_p,
                                        float* __restrict__ out) {
  float acc = 0.0f;
  for (int i = threadIdx.x; i < n; i += blockDim.x) acc += partials[i];
  float v = block_reduce(acc);
  if (threadIdx.x == 0) out[0] = v * (float)(*kc_p);
}

extern "C" void kernel_launch(void* const* d_in, const int* in_sizes, int n_in,
                              void* d_out, int out_size, void* d_ws, size_t ws_size,
                              hipStream_t stream) {
  const float* coords  = (const float*)d_in[0];   // [100000][3] f32
  const int*   pairs   = (const int*)d_in[1];     // [8000000][2] i32
  const float* box     = (const float*)d_in[2];   // [3][3] f32 (diagonal)
  const float* charges = (const float*)d_in[3];   // [100000] f32
  const int*   kc_p    = (const int*)d_in[4];     // scalar 139
  const int*   cut_p   = (const int*)d_in[5];     // scalar 10

  const int n_pairs = in_sizes[1] / 2;

  int nblocks = NBLOCKS;
  if ((size_t)nblocks * sizeof(float) > ws_size) {
    nblocks = (int)(ws_size / sizeof(float));
    if (nblocks < 1) nblocks = 1;
  }
  float* partials = (float*)d_ws;

  coulomb_pairs_kernel<<<nblocks, TPB, 0, stream>>>(
      coords, pairs, box, charges, cut_p, partials, n_pairs);
  coulomb_finalize_kernel<<<1, TPB, 0, stream>>>(
      partials, nblocks, kc_p, (float*)d_out);
}